// MjCambrianOptics_63591285785075
// MI455X (gfx1250) — compile-verified
//
#include <hip/hip_runtime.h>
#include <hip/hip_bf16.h>
#include <math.h>

// ---------------------------------------------------------------------------
// MjCambrianOptics on gfx1250 (MI455X):
//   depth mean -> complex fields -> FFTs as f32 WMMA DFT-GEMMs -> PSF ->
//   depthwise 255x255 conv (LDS row-tiled, branchless) -> crop 160 -> clip
// ---------------------------------------------------------------------------

#define MRES   255            // PAD_RES
#define NP     256            // padded GEMM dim (16 | 256)
#define NB     (NP * NP)      // floats per padded buffer
#define NPIX   (MRES * MRES)  // 65025
#define ORES   160            // output RES
#define TWO_PI 6.28318530717958647692f

typedef float v2f __attribute__((ext_vector_type(2)));
typedef float v8f __attribute__((ext_vector_type(8)));

// ---------------- scalar helpers ----------------

__global__ void k_zero_scalars(float* scal) {
    if (threadIdx.x < 8) scal[threadIdx.x] = 0.0f;
}

__global__ void k_reduce_depth(const float* __restrict__ depth, float* __restrict__ scal) {
    __shared__ float sm[256];
    int tid = threadIdx.x;
    float s = 0.0f;
    for (int i = blockIdx.x * 256 + tid; i < NPIX; i += gridDim.x * 256) s += depth[i];
    sm[tid] = s;
    __syncthreads();
    for (int st = 128; st > 0; st >>= 1) {
        if (tid < st) sm[tid] += sm[tid + st];
        __syncthreads();
    }
    if (tid == 0) atomicAdd(&scal[0], sm[0]);
}

// ---------------- DFT matrix F[j][k] = exp(-2*pi*i*j*k/255), zero padded ----

__global__ void k_build_F(float* __restrict__ Fr, float* __restrict__ Fi) {
    int idx = blockIdx.x * 256 + threadIdx.x;   // 65536 total
    int j = idx >> 8, k = idx & 255;
    float cr = 0.0f, ci = 0.0f;
    if (j < MRES && k < MRES) {
        int t = (j * k) % MRES;                 // keep the phase argument small
        float ang = -TWO_PI * (float)t * (1.0f / 255.0f);
        cr = cosf(ang);
        ci = sinf(ang);
    }
    Fr[idx] = cr;
    Fi[idx] = ci;
}

// ---------------- complex fields: mode 0 = u2 (pupil), mode 1 = H ----------

__global__ void k_fields(const float* __restrict__ scal,
                         float* __restrict__ outR, float* __restrict__ outI,
                         float wl, int mode) {
    int idx = blockIdx.x * 256 + threadIdx.x;
    int i = idx >> 8, j = idx & 255;
    float re = 0.0f, im = 0.0f;
    if (i < MRES && j < MRES) {
        float md = scal[0] * (1.0f / (float)NPIX);  // mean depth
        if (mode == 0) {
            // x1,y1 = linspace(-0.001, 0.001, 255)
            float x = -0.001f + (float)i * (0.002f / 254.0f);
            float y = -0.001f + (float)j * (0.002f / 254.0f);
            float r2 = x * x + y * y;
            const float ar = 0.001f * 0.1f + 1e-7f;           // aperture radius
            float A = (sqrtf(r2) / ar <= 1.0f) ? 1.0f : 0.0f;
            float ph = (TWO_PI / wl) * sqrtf(r2 + md * md);   // u1 = exp(i*ph)
            re = A * cosf(ph);
            im = A * sinf(ph);
        } else {
            const float fmax = (float)MRES / (2.0f * 0.002f); // 63750 (ref uses dx twice)
            float fx = -fmax + (float)i * (2.0f * fmax / 254.0f);
            float fy = -fmax + (float)j * (2.0f * fmax / 254.0f);
            float hv = (sqrtf(fx * fx + fy * fy) < 1.0f / wl) ? 1.0f : 0.0f;
            float a = 1.0f - (wl * fx) * (wl * fx) - (wl * fy) * (wl * fy);
            float fxfy = sqrtf(fmaxf(a, 0.0f));
            float ph = (TWO_PI / wl) * md * fxfy;
            re = hv * cosf(ph);
            im = hv * sinf(ph);
        }
    }
    outR[idx] = re;
    outI[idx] = im;
}

// ---------------- complex GEMM via V_WMMA_F32_16X16X4_F32 -------------------
// C = alpha * (Ar + i*sA*Ai) x (Br + i*sB*Bi), all 256x256 row-major.
// One wave computes a 16x32 strip (two 16x16 N-tiles sharing the A fragment):
// 8 independent WMMA accumulator chains per K-step hide the WMMA RAW latency.
// Grid: 16 M-tiles x 8 N-groups = 128 waves = 16 blocks x 8 waves.
// A fragment (16x4 f32): lanes 0-15 K={0,1}, lanes 16-31 K={2,3} (ISA 7.12.2)
// C/D layout: VGPR r -> M = r + 8*(lane>=16), N = lane&15.

__global__ void __launch_bounds__(256)
k_cgemm(const float* __restrict__ Ar, const float* __restrict__ Ai,
        const float* __restrict__ Br, const float* __restrict__ Bi,
        float* __restrict__ Cr, float* __restrict__ Ci,
        float sA, float sB, float alpha) {
    int tid  = threadIdx.x;
    int wave = tid >> 5;
    int lane = tid & 31;
    int wid  = blockIdx.x * 8 + wave;       // 0..127
    int tm   = (wid >> 3) * 16;             // 16 M-tiles
    int tn   = (wid & 7) * 32;              // 8 N-groups of 32 cols
    int half = lane >> 4;                   // 0: K pair {0,1}, 1: K pair {2,3}
    int l16  = lane & 15;
    int arow = tm + l16;
    int bc0  = tn + l16;                    // N-tile 0 column
    int bc1  = tn + 16 + l16;               // N-tile 1 column

    v8f rr0 = {0.f,0.f,0.f,0.f,0.f,0.f,0.f,0.f};
    v8f ri0 = rr0, ir0 = rr0, ii0 = rr0;
    v8f rr1 = rr0, ri1 = rr0, ir1 = rr0, ii1 = rr0;

    for (int k0 = 0; k0 < NP; k0 += 4) {
        int ka = k0 + 2 * half;
        v2f a_r = *(const v2f*)(Ar + arow * NP + ka);   // A[m][ka], A[m][ka+1]
        v2f a_i = *(const v2f*)(Ai + arow * NP + ka);
        v2f br0, bi0, br1, bi1;
        br0.x = Br[ ka      * NP + bc0];
        br0.y = Br[(ka + 1) * NP + bc0];
        bi0.x = Bi[ ka      * NP + bc0];
        bi0.y = Bi[(ka + 1) * NP + bc0];
        br1.x = Br[ ka      * NP + bc1];
        br1.y = Br[(ka + 1) * NP + bc1];
        bi1.x = Bi[ ka      * NP + bc1];
        bi1.y = Bi[(ka + 1) * NP + bc1];

        rr0 = __builtin_amdgcn_wmma_f32_16x16x4_f32(false, a_r, false, br0, (short)0, rr0, false, false);
        ri0 = __builtin_amdgcn_wmma_f32_16x16x4_f32(false, a_r, false, bi0, (short)0, ri0, false, false);
        ir0 = __builtin_amdgcn_wmma_f32_16x16x4_f32(false, a_i, false, br0, (short)0, ir0, false, false);
        ii0 = __builtin_amdgcn_wmma_f32_16x16x4_f32(false, a_i, false, bi0, (short)0, ii0, false, false);
        rr1 = __builtin_amdgcn_wmma_f32_16x16x4_f32(false, a_r, false, br1, (short)0, rr1, false, false);
        ri1 = __builtin_amdgcn_wmma_f32_16x16x4_f32(false, a_r, false, bi1, (short)0, ri1, false, false);
        ir1 = __builtin_amdgcn_wmma_f32_16x16x4_f32(false, a_i, false, br1, (short)0, ir1, false, false);
        ii1 = __builtin_amdgcn_wmma_f32_16x16x4_f32(false, a_i, false, bi1, (short)0, ii1, false, false);
    }

    float ss = sA * sB;
    #pragma unroll
    for (int r = 0; r < 8; ++r) {
        int m = tm + r + 8 * half;
        Cr[m * NP + bc0] = alpha * (rr0[r] - ss * ii0[r]);
        Ci[m * NP + bc0] = alpha * (sB * ri0[r] + sA * ir0[r]);
        Cr[m * NP + bc1] = alpha * (rr1[r] - ss * ii1[r]);
        Ci[m * NP + bc1] = alpha * (sB * ri1[r] + sA * ir1[r]);
    }
}

// ---------------- elementwise complex multiply ------------------------------

__global__ void k_cmul(const float* __restrict__ Xr, const float* __restrict__ Xi,
                       const float* __restrict__ Yr, const float* __restrict__ Yi,
                       float* __restrict__ Pr, float* __restrict__ Pi) {
    int idx = blockIdx.x * 256 + threadIdx.x;
    float xr = Xr[idx], xi = Xi[idx], yr = Yr[idx], yi = Yi[idx];
    Pr[idx] = xr * yr - xi * yi;
    Pi[idx] = xr * yi + xi * yr;
}

// ---------------- PSF: complex square, complex sum, normalize ---------------

__global__ void k_psf_reduce(const float* __restrict__ u3r, const float* __restrict__ u3i,
                             float* __restrict__ scal, int c) {
    __shared__ float smr[256], smi[256];
    int tid = threadIdx.x;
    float sr = 0.0f, si = 0.0f;
    for (int i = blockIdx.x * 256 + tid; i < NPIX; i += gridDim.x * 256) {
        int row = i / MRES, col = i - row * MRES;
        float ur = u3r[row * NP + col], ui = u3i[row * NP + col];
        sr += ur * ur - ui * ui;      // Re(u3^2)
        si += 2.0f * ur * ui;         // Im(u3^2)
    }
    smr[tid] = sr; smi[tid] = si;
    __syncthreads();
    for (int st = 128; st > 0; st >>= 1) {
        if (tid < st) { smr[tid] += smr[tid + st]; smi[tid] += smi[tid + st]; }
        __syncthreads();
    }
    if (tid == 0) {
        atomicAdd(&scal[1 + c], smr[0]);
        atomicAdd(&scal[4 + c], smi[0]);
    }
}

__global__ void k_psf_norm(const float* __restrict__ u3r, const float* __restrict__ u3i,
                           const float* __restrict__ scal, float* __restrict__ psf, int c) {
    int i = blockIdx.x * 256 + threadIdx.x;
    if (i >= NPIX) return;
    int row = i / MRES, col = i - row * MRES;
    float ur = u3r[row * NP + col], ui = u3i[row * NP + col];
    float pr = ur * ur - ui * ui;
    float pi = 2.0f * ur * ui;
    float Sr = scal[1 + c] + 1e-7f;               // complex sum + 1e-7
    float Si = scal[4 + c];
    // Re( (pr + i*pi) / (Sr + i*Si) )
    psf[c * NPIX + i] = (pr * Sr + pi * Si) / (Sr * Sr + Si * Si);
}

// ---------------- depthwise 255x255 conv, SAME, crop 160x160, clip ----------
// One block per (channel, output row). LDS-staged zero-padded image row +
// psf row; the pad makes the inner 255-tap FMA loop branchless so the
// compiler can unroll and merge LDS loads.

#define PADW 127
#define SROWW (MRES + 2 * PADW)   // 509

__global__ void __launch_bounds__(160)
k_conv(const float* __restrict__ img, const float* __restrict__ psf,
       float* __restrict__ out) {
    __shared__ float srow[SROWW];
    __shared__ float sker[MRES];
    int b   = blockIdx.x;        // 0..479
    int c   = b / ORES;
    int oy  = b - c * ORES;
    int tid = threadIdx.x;       // 0..159 -> output x
    int oi  = oy + 47;           // row in full 255x255 output

    // zero the pad regions once (interior is overwritten every di)
    for (int x = tid; x < SROWW; x += 160)
        if (x < PADW || x >= PADW + MRES) srow[x] = 0.0f;
    __syncthreads();

    float acc = 0.0f;
    for (int di = 0; di < MRES; ++di) {
        int ai = oi + di - 127;                 // uniform across block
        if (ai < 0 || ai >= MRES) continue;
        for (int x = tid; x < MRES; x += 160) {
            srow[PADW + x] = img[(ai * MRES + x) * 3 + c];
            sker[x]        = psf[c * NPIX + di * MRES + x];
        }
        __syncthreads();
        // out col = tid+47; image col = (tid+47) + dj - 127 -> srow[tid+47+dj]
        const float* rp = &srow[tid + 47];
        #pragma unroll 5
        for (int dj = 0; dj < MRES; ++dj)
            acc += sker[dj] * rp[dj];
        __syncthreads();
    }
    out[(oy * ORES + tid) * 3 + c] = fminf(fmaxf(acc, 0.0f), 1.0f);
}

// ---------------------------------------------------------------------------

extern "C" void kernel_launch(void* const* d_in, const int* in_sizes, int n_in,
                              void* d_out, int out_size, void* d_ws, size_t ws_size,
                              hipStream_t stream) {
    const float* img   = (const float*)d_in[0];  // 255*255*3
    const float* depth = (const float*)d_in[1];  // 255*255
    float* out = (float*)d_out;                  // 160*160*3 f32

    float* ws   = (float*)d_ws;
    float* scal = ws;                 // [0]=depth sum, [1..3]=Sr, [4..6]=Si
    float* Fr   = ws + 64;
    float* Fi   = Fr  + NB;
    float* B0r  = Fi  + NB;
    float* B0i  = B0r + NB;
    float* B1r  = B0i + NB;
    float* B1i  = B1r + NB;
    float* B2r  = B1i + NB;
    float* B2i  = B2r + NB;
    float* psf  = B2i + NB;           // 3 * 65025

    const float invNN = 1.0f / ((float)NPIX);    // 1/(255*255) ifft scale
    const float wls[3] = {6.1e-07f, 5.3e-07f, 4.7e-07f};

    k_zero_scalars<<<1, 32, 0, stream>>>(scal);
    k_reduce_depth<<<64, 256, 0, stream>>>(depth, scal);
    k_build_F<<<NB / 256, 256, 0, stream>>>(Fr, Fi);

    for (int c = 0; c < 3; ++c) {
        float wl = wls[c];
        // ---- fft2(u2) = F * u2 * F  -> B2
        k_fields<<<NB / 256, 256, 0, stream>>>(scal, B0r, B0i, wl, 0);
        k_cgemm<<<16, 256, 0, stream>>>(Fr, Fi, B0r, B0i, B1r, B1i, 1.f, 1.f, 1.f);
        k_cgemm<<<16, 256, 0, stream>>>(B1r, B1i, Fr, Fi, B2r, B2i, 1.f, 1.f, 1.f);
        // ---- fft2(H) -> B0
        k_fields<<<NB / 256, 256, 0, stream>>>(scal, B0r, B0i, wl, 1);
        k_cgemm<<<16, 256, 0, stream>>>(Fr, Fi, B0r, B0i, B1r, B1i, 1.f, 1.f, 1.f);
        k_cgemm<<<16, 256, 0, stream>>>(B1r, B1i, Fr, Fi, B0r, B0i, 1.f, 1.f, 1.f);
        // ---- pointwise product -> B1
        k_cmul<<<NB / 256, 256, 0, stream>>>(B0r, B0i, B2r, B2i, B1r, B1i);
        // ---- ifft2 = (1/N^2) conj(F) * P * conj(F) -> B0 (= u3)
        k_cgemm<<<16, 256, 0, stream>>>(Fr, Fi, B1r, B1i, B2r, B2i, -1.f, 1.f, 1.f);
        k_cgemm<<<16, 256, 0, stream>>>(B2r, B2i, Fr, Fi, B0r, B0i, 1.f, -1.f, invNN);
        // ---- PSF normalization
        k_psf_reduce<<<128, 256, 0, stream>>>(B0r, B0i, scal, c);
        k_psf_norm<<<(NPIX + 255) / 256, 256, 0, stream>>>(B0r, B0i, scal, psf, c);
    }

    k_conv<<<3 * ORES, 160, 0, stream>>>(img, psf, out);
}